// HQTCN_70652212019648
// MI455X (gfx1250) — compile-verified
//
#include <hip/hip_runtime.h>

// ---------------------------------------------------------------------------
// HQTCN on MI455X (gfx1250, wave32):
//  The quantum circuit is sample-independent -> it is ONE fixed 1024x1024
//  complex unitary U.  We build U once (basis-column evolution in LDS), then
//  per-sample evolution is a dense GEMM  Z = U * Psi  (Psi is real), done with
//  v_wmma_f32_16x16x4_f32.  Operands are pair-packed in memory so every WMMA
//  fragment is a single aligned global_load_b64.
// ---------------------------------------------------------------------------

typedef __attribute__((ext_vector_type(2))) float v2f;
typedef __attribute__((ext_vector_type(8))) float v8f;

#define NQ     10
#define NSTATE 1024          // 2^10
#define S_TOT  1936          // 16 * 121 windows
#define S_TILES 121
#define K_FC   256           // C*KERNEL = 32*8
#define T_LEN  128
#define C_LEN  32
#define NWIN   121
#define PAIRS  (NSTATE/2)    // 512 packed K-pairs
#define AP     (2*NSTATE)    // 2048: elements per pair-slab of U
#define BP     (2*S_TOT)     // 3872: elements per pair-slab of Psi

#ifndef __has_builtin
#define __has_builtin(x) 0
#endif
#if __has_builtin(__builtin_amdgcn_global_load_async_to_lds_b32)
#define USE_ASYNC_LDS 1
typedef __attribute__((address_space(1))) int* gint_p;
typedef __attribute__((address_space(3))) int* lint_p;
#else
#define USE_ASYNC_LDS 0
#endif

// ---------------- complex helpers ----------------
struct cpx { float re, im; };
__device__ inline cpx cmk(float r, float i){ cpx c; c.re=r; c.im=i; return c; }
__device__ inline cpx cmul(cpx a, cpx b){ return cmk(a.re*b.re - a.im*b.im, a.re*b.im + a.im*b.re); }
__device__ inline cpx cadd(cpx a, cpx b){ return cmk(a.re+b.re, a.im+b.im); }
__device__ inline cpx cexpi(float t){ return cmk(cosf(t), sinf(t)); }
__device__ inline cpx cscale(cpx a, float f){ return cmk(a.re*f, a.im*f); }

__device__ inline unsigned insert_bit(unsigned x, int p, unsigned v){
    unsigned lo = x & ((1u << p) - 1u);
    return ((x >> p) << (p + 1)) | (v << p) | lo;
}

// ======================== kernel 0: zero output ============================
__global__ void zero_out_k(float* out, int n){
    int i = blockIdx.x * 256 + threadIdx.x;
    if (i < n) out[i] = 0.0f;
}

// ============ kernel 1: angles GEMM (WMMA f32 16x16x4) + cos/sin ===========
// angles[s,e] = sum_f flat[s,f]*fc_w[e,f] + fc_b[e];  flat[s, c*8+j] = x[b,c,w+j]
// M = 1936 (121 tiles of 16), N = 16 (10 valid), K = 256.
// fc_w is staged zero-padded into LDS -> no divergence in the WMMA loop.
__global__ __launch_bounds__(32) void angles_wmma_k(const float* __restrict__ x,
                                                    const float* __restrict__ fcw,
                                                    const float* __restrict__ fcb,
                                                    float* __restrict__ ch,
                                                    float* __restrict__ sh){
    __shared__ float lfw[16 * K_FC];     // 16 KB, rows 10..15 are zero

#if USE_ASYNC_LDS
    for (int t = threadIdx.x; t < NQ * K_FC; t += 32) {
        __builtin_amdgcn_global_load_async_to_lds_b32(
            (gint_p)(unsigned long long)(const void*)(fcw + t),
            (lint_p)(lfw + t), 0, 0);
    }
    for (int t = NQ * K_FC + (int)threadIdx.x; t < 16 * K_FC; t += 32) lfw[t] = 0.0f;
    asm volatile("s_wait_asynccnt 0x0" ::: "memory");
#else
    for (int t = threadIdx.x; t < 16 * K_FC; t += 32) {
        int e = t >> 8, k = t & 255;
        lfw[t] = (e < NQ) ? fcw[e * K_FC + k] : 0.0f;
    }
#endif
    __syncthreads();

    const int st = blockIdx.x;           // M tile 0..120
    const int l  = threadIdx.x;          // lane
    const int m  = l & 15;
    const int hi = l >> 4;               // lanes 16-31 hold K+2,K+3 (ISA 16x4 A layout)
    const int s  = st * 16 + m;
    const int b  = s / NWIN;
    const int w  = s % NWIN;
    const float* xb = x + b * C_LEN * T_LEN;
    const float* lb = lfw + m * K_FC;    // this lane's fc_w row (zero-padded)

    v8f acc = {};
    for (int k0 = 0; k0 < K_FC; k0 += 4) {
        const int ka = k0 + 2 * hi;      // even -> j0 = ka&7 is even -> contiguous pair
        const int c0 = ka >> 3, j0 = ka & 7;
        const float* ap = xb + c0 * T_LEN + w + j0;
        v2f a;  a.x = ap[0];  a.y = ap[1];
        v2f bb; bb.x = lb[ka]; bb.y = lb[ka + 1];   // ds_load
        acc = __builtin_amdgcn_wmma_f32_16x16x4_f32(false, a, false, bb,
                                                    (short)0, acc, false, false);
    }
    // C layout: VGPR v -> row (v + 8*hi), col = lane%16
    const int e = m;
    if (e < NQ) {
#pragma unroll
        for (int v = 0; v < 8; ++v) {
            int ss  = st * 16 + v + hi * 8;
            float h = 0.5f * (acc[v] + fcb[e]);
            ch[ss * NQ + e] = cosf(h);
            sh[ss * NQ + e] = sinf(h);
        }
    }
}

// ============== kernel 2: initial states, pair-packed ======================
// Psi_pk[p*BP + 2*s + c] = amp(state 2p+c, sample s); qubit9 (LSB) split last.
__global__ __launch_bounds__(256) void init_psi_k(const float* __restrict__ ch,
                                                  const float* __restrict__ sh,
                                                  float* __restrict__ Psi_pk){
    const int p = blockIdx.x;            // pair index 0..511
    for (int s = threadIdx.x; s < S_TOT; s += 256) {
        float prod = 1.0f;
#pragma unroll
        for (int q = 0; q < 9; ++q) {
            bool bit = (p >> (8 - q)) & 1;
            prod *= bit ? sh[s * NQ + q] : ch[s * NQ + q];
        }
        v2f v;
        v.x = prod * ch[s * NQ + 9];     // state 2p   (qubit9 = 0)
        v.y = prod * sh[s * NQ + 9];     // state 2p+1 (qubit9 = 1)
        *(v2f*)(Psi_pk + p * BP + 2 * s) = v;
    }
}

// ============ kernel 3: build U by evolving basis columns in LDS ===========
// One block per basis column j. Pair-packed store:
//   Upk[(j>>1)*AP + 2*i + (j&1)] = U[i, j]

__device__ inline void g_1q(float* sre, float* sim, int tid, int q,
                            cpx a00, cpx a01, cpx a10, cpx a11){
    const int p = 9 - q;
    const int msk = 1 << p;
    __syncthreads();
#pragma unroll 2
    for (int t = tid; t < 512; t += 256) {
        int i0 = (int)insert_bit((unsigned)t, p, 0u);
        int i1 = i0 | msk;
        cpx s0 = cmk(sre[i0], sim[i0]);
        cpx s1 = cmk(sre[i1], sim[i1]);
        cpx n0 = cadd(cmul(a00, s0), cmul(a01, s1));
        cpx n1 = cadd(cmul(a10, s0), cmul(a11, s1));
        sre[i0] = n0.re; sim[i0] = n0.im;
        sre[i1] = n1.re; sim[i1] = n1.im;
    }
}

__device__ inline void g_u3(float* sre, float* sim, int tid, int q,
                            float th, float ph, float dl){
    float ct = cosf(0.5f * th), st = sinf(0.5f * th);
    cpx a00 = cmk(ct, 0.0f);
    cpx a01 = cscale(cexpi(dl), -st);
    cpx a10 = cscale(cexpi(ph),  st);
    cpx a11 = cscale(cexpi(ph + dl), ct);
    g_1q(sre, sim, tid, q, a00, a01, a10, a11);
}

__device__ inline void g_zz(float* sre, float* sim, int tid, int q1, int q2, float t){
    float c = cosf(0.5f * t), sn = sinf(0.5f * t);
    const int m1 = 1 << (9 - q1), m2 = 1 << (9 - q2);
    __syncthreads();
#pragma unroll 4
    for (int i = tid; i < NSTATE; i += 256) {
        bool b1 = (i & m1) != 0, b2 = (i & m2) != 0;
        float ph_im = (b1 == b2) ? -sn : sn;   // e^{-it/2} on 00/11, e^{+it/2} else
        cpx s = cmk(sre[i], sim[i]);
        cpx n = cmul(cmk(c, ph_im), s);
        sre[i] = n.re; sim[i] = n.im;
    }
}

// XX: new[i] = c*s[i] - i*sn*s[i^mask]          (f = -sn for both pair members)
// YY: pair (0,b2)<->(1,~b2): f = +sn if b2==0 else -sn
__device__ inline void g_xxyy(float* sre, float* sim, int tid, int q1, int q2,
                              float t, bool yy){
    float c = cosf(0.5f * t), sn = sinf(0.5f * t);
    const int p1 = 9 - q1;
    const int m1 = 1 << (9 - q1), m2 = 1 << (9 - q2);
    __syncthreads();
#pragma unroll 2
    for (int tt = tid; tt < 512; tt += 256) {
        int i0 = (int)insert_bit((unsigned)tt, p1, 0u);   // bit q1 == 0
        int j  = i0 ^ (m1 | m2);
        float f = yy ? (((i0 & m2) == 0) ? sn : -sn) : -sn;
        cpx a = cmk(sre[i0], sim[i0]);
        cpx b = cmk(sre[j],  sim[j]);
        // n = c*self + i*f*partner ;  i*f*(x+iy) = (-f*y, f*x)
        cpx n0 = cmk(c * a.re - f * b.im, c * a.im + f * b.re);
        cpx n1 = cmk(c * b.re - f * a.im, c * b.im + f * a.re);
        sre[i0] = n0.re; sim[i0] = n0.im;
        sre[j]  = n1.re; sim[j]  = n1.im;
    }
}

__device__ inline void g_crot(float* sre, float* sim, int tid, int ctrl, int tgt,
                              float ph, float th, float om){
    // reference crot(phi, theta, omega): r applied on tgt when ctrl bit == 1
    float ct = cosf(0.5f * th), st = sinf(0.5f * th);
    cpx r00 = cscale(cexpi(-0.5f * (ph + om)), ct);
    cpx r01 = cscale(cexpi( 0.5f * (ph - om)), -st);
    cpx r10 = cscale(cexpi(-0.5f * (ph - om)), st);
    cpx r11 = cscale(cexpi( 0.5f * (ph + om)), ct);
    const int pc = 9 - ctrl, pt = 9 - tgt;
    const int plo = pc < pt ? pc : pt;
    const int phi_ = pc < pt ? pt : pc;
    __syncthreads();
    for (int t = tid; t < 256; t += 256) {
        unsigned xI = insert_bit((unsigned)t, plo, 0u);
        xI = insert_bit(xI, phi_, 0u);
        int i0 = (int)(xI | (1u << pc));   // ctrl=1, tgt=0
        int i1 = i0 | (1 << pt);
        cpx s0 = cmk(sre[i0], sim[i0]);
        cpx s1 = cmk(sre[i1], sim[i1]);
        cpx n0 = cadd(cmul(r00, s0), cmul(r01, s1));
        cpx n1 = cadd(cmul(r10, s0), cmul(r11, s1));
        sre[i0] = n0.re; sim[i0] = n0.im;
        sre[i1] = n1.re; sim[i1] = n1.im;
    }
}

__global__ __launch_bounds__(256) void evolve_cols_k(const float* __restrict__ convp,
                                                     const float* __restrict__ poolp,
                                                     float* __restrict__ Upk_re,
                                                     float* __restrict__ Upk_im){
    __shared__ float sre[NSTATE];
    __shared__ float sim[NSTATE];
    const int j   = blockIdx.x;          // basis column
    const int tid = threadIdx.x;

    for (int i = tid; i < NSTATE; i += 256) { sre[i] = 0.0f; sim[i] = 0.0f; }
    __syncthreads();
    if (tid == 0) sre[j] = 1.0f;

    int wires[NQ] = {0,1,2,3,4,5,6,7,8,9};
    int nw = NQ;
    for (int layer = 0; layer < 3; ++layer) {
        const float* W = convp + layer * (NQ * 15);
        const float* P = poolp + layer * (5 * 3);
        for (int p = 0; p < 2; ++p) {
            for (int indx = 0; indx < nw - 1; ++indx) {
                if ((indx & 1) != p) continue;
                int w1 = wires[indx], w2 = wires[indx + 1];
                const float* Wa = W + indx * 15;
                const float* Wb = W + (indx + 1) * 15;
                g_u3(sre, sim, tid, w1, Wa[0], Wa[1], Wa[2]);
                g_u3(sre, sim, tid, w2, Wb[3], Wb[4], Wb[5]);
                g_zz(sre, sim, tid, w1, w2, Wa[6]);
                g_xxyy(sre, sim, tid, w1, w2, Wa[7], true);   // YY
                g_xxyy(sre, sim, tid, w1, w2, Wa[8], false);  // XX
                g_u3(sre, sim, tid, w1, Wa[9], Wa[10], Wa[11]);
                g_u3(sre, sim, tid, w2, Wb[12], Wb[13], Wb[14]);
            }
        }
        for (int jj = 0; jj < nw / 2; ++jj) {
            int ctrl = wires[2 * jj + 1], tgt = wires[2 * jj];
            g_crot(sre, sim, tid, ctrl, tgt, P[jj*3+0], P[jj*3+1], P[jj*3+2]);
        }
        int nn = (nw + 1) / 2;
        for (int t = 0; t < nn; ++t) wires[t] = wires[2 * t];
        nw = nn;
    }
    __syncthreads();
    const int slab = (j >> 1) * AP + (j & 1);
    for (int i = tid; i < NSTATE; i += 256) {
        Upk_re[slab + 2 * i] = sre[i];
        Upk_im[slab + 2 * i] = sim[i];
    }
}

// ====== kernel 4: Z = U * Psi (two real WMMA GEMMs) + fused readout ========
// Each wave: 2 i-tiles (32 rows) x 1 s-tile, sharing the B fragment.
// All fragments are aligned b64 loads from the pair-packed layouts.
__global__ __launch_bounds__(256) void zgemm_wmma_k(const float* __restrict__ Upk_re,
                                                    const float* __restrict__ Upk_im,
                                                    const float* __restrict__ Psi_pk,
                                                    float* __restrict__ out){
    const int wid = threadIdx.x >> 5;
    const int l   = threadIdx.x & 31;
    const int wt  = blockIdx.x * 8 + wid;    // 3872 wave-tiles = 32 i-groups x 121
    const int ig  = wt & 31;                 // 32-row i-group
    const int st  = wt >> 5;                 // s tile
    const int m   = l & 15;
    const int hi  = l >> 4;                  // K split per ISA 16x4 A layout
    const int i0  = ig * 32 + m;
    const int s   = st * 16 + m;

    const float* par = Upk_re + 2 * i0 + hi * AP;
    const float* pai = Upk_im + 2 * i0 + hi * AP;
    const float* pb  = Psi_pk + 2 * s  + hi * BP;

    v8f c0re = {}, c0im = {}, c1re = {}, c1im = {};
#pragma unroll 4
    for (int k0 = 0; k0 < NSTATE; k0 += 4) {
        const int oa = (k0 >> 1) * AP;       // pair slab offset (A)
        const int ob = (k0 >> 1) * BP;       // pair slab offset (B)
        __builtin_prefetch(par + oa + 8 * AP, 0, 1);
        __builtin_prefetch(pai + oa + 8 * AP, 0, 1);
        v2f a0r = *(const v2f*)(par + oa);
        v2f a0i = *(const v2f*)(pai + oa);
        v2f a1r = *(const v2f*)(par + oa + 32);   // rows i0+16
        v2f a1i = *(const v2f*)(pai + oa + 32);
        v2f bb  = *(const v2f*)(pb  + ob);
        c0re = __builtin_amdgcn_wmma_f32_16x16x4_f32(false, a0r, false, bb, (short)0, c0re, false, false);
        c0im = __builtin_amdgcn_wmma_f32_16x16x4_f32(false, a0i, false, bb, (short)0, c0im, false, false);
        c1re = __builtin_amdgcn_wmma_f32_16x16x4_f32(false, a1r, false, bb, (short)0, c1re, false, false);
        c1im = __builtin_amdgcn_wmma_f32_16x16x4_f32(false, a1i, false, bb, (short)0, c1im, false, false);
    }
    // sign is uniform per 32-row group: rows [ig*32, ig*32+31], boundary at 512
    const float sgn = (ig < 16) ? 1.0f : -1.0f;
    float acc = 0.0f;
#pragma unroll
    for (int v = 0; v < 8; ++v) {
        acc += c0re[v] * c0re[v] + c0im[v] * c0im[v];
        acc += c1re[v] * c1re[v] + c1im[v] * c1im[v];
    }
    acc *= sgn;
    acc += __shfl_xor(acc, 16, 32);          // combine lane halves (same sample)
    if (hi == 0) atomicAdd(out + s, acc);    // global_atomic_add_f32
}

// ============================== launcher ===================================
extern "C" void kernel_launch(void* const* d_in, const int* in_sizes, int n_in,
                              void* d_out, int out_size, void* d_ws, size_t ws_size,
                              hipStream_t stream) {
    const float* x     = (const float*)d_in[0];   // (16,32,128)
    const float* fcw   = (const float*)d_in[1];   // (10,256)
    const float* fcb   = (const float*)d_in[2];   // (10,)
    const float* convp = (const float*)d_in[3];   // (3,10,15)
    const float* poolp = (const float*)d_in[4];   // (3,5,3)
    float* out = (float*)d_out;                   // (16,121) = 1936

    float* ws     = (float*)d_ws;
    float* ch     = ws;                           // S_TOT*10
    float* sh     = ch + S_TOT * NQ;              // S_TOT*10
    float* Psi_pk = sh + S_TOT * NQ;              // 512*3872
    float* Ure    = Psi_pk + PAIRS * BP;          // 512*2048
    float* Uim    = Ure + PAIRS * AP;             // 512*2048
    (void)in_sizes; (void)n_in; (void)ws_size;

    zero_out_k   <<<(out_size + 255) / 256, 256, 0, stream>>>(out, out_size);
    angles_wmma_k<<<S_TILES, 32, 0, stream>>>(x, fcw, fcb, ch, sh);
    init_psi_k   <<<PAIRS, 256, 0, stream>>>(ch, sh, Psi_pk);
    evolve_cols_k<<<NSTATE, 256, 0, stream>>>(convp, poolp, Ure, Uim);
    zgemm_wmma_k <<<(32 * S_TILES) / 8, 256, 0, stream>>>(Ure, Uim, Psi_pk, out);
}